// EdgeTypeSpecGCNLayerLocal_86801289052298
// MI455X (gfx1250) — compile-verified
//
#include <hip/hip_runtime.h>
#include <stdint.h>

#define NNODES 50000
#define NEDGES 600000
#define NSUB   4
#define DIM    128
#define NTILES (NNODES / 16)          // 3125, exact
#define KEEP_SCALE (1.0f / 0.9f)

typedef __attribute__((ext_vector_type(2))) float v2f;
typedef __attribute__((ext_vector_type(8))) float v8f;

// Deterministic per-(subtype, element) Bernoulli(0.9) dropout mask * 1/(1-p).
__device__ __forceinline__ float drop_mask(uint32_t idx, uint32_t s) {
    uint32_t h = idx ^ (s * 0x9E3779B9u + 0x85EBCA6Bu);
    h *= 0x7FEB352Du; h ^= h >> 15; h *= 0x846CA68Bu; h ^= h >> 13;
    return ((h & 0xFFFFFu) < 943718u) ? KEEP_SCALE : 0.0f;  // 0.9 * 2^20
}

__global__ void __launch_bounds__(256)
k_zero4(float* __restrict__ p, int n4) {
    int i = blockIdx.x * blockDim.x + threadIdx.x;
    if (i < n4) ((float4*)p)[i] = make_float4(0.f, 0.f, 0.f, 0.f);
}

__global__ void __launch_bounds__(256)
k_degree(const int* __restrict__ dst, float* __restrict__ deg, int e) {
    int i = blockIdx.x * blockDim.x + threadIdx.x;
    if (i < e) atomicAdd(&deg[dst[i]], 1.0f);
}

__global__ void __launch_bounds__(256)
k_dinv(float* __restrict__ deg, int n) {   // in place: deg -> dinv, +1 self loop
    int i = blockIdx.x * blockDim.x + threadIdx.x;
    if (i < n) deg[i] = rsqrtf(deg[i] + 1.0f);
}

// h'[row, :] = (dropout_s(x[row,:]) @ W) * dinv[row]
// One wave32 computes a 16-row x 128-col strip with V_WMMA_F32_16X16X4_F32.
__global__ void __launch_bounds__(128)
k_gemm_wmma(const float* __restrict__ x, const float* __restrict__ W,
            const float* __restrict__ dinv, float* __restrict__ h, int s) {
    __shared__ float sW[DIM * DIM];     // 64 KB of the 320 KB/WGP LDS
    const int tid = threadIdx.x;
    for (int i = tid; i < DIM * DIM / 4; i += 128)
        ((float4*)sW)[i] = ((const float4*)W)[i];
    __syncthreads();

    const int wave = tid >> 5;
    const int lane = tid & 31;
    const int half = lane >> 4;
    const int l16  = lane & 15;
    const int tile = blockIdx.x * 4 + wave;
    if (tile >= NTILES) return;

    const int rowA = tile * 16 + l16;               // A-matrix: lane = row M
    const float* xrow = x + (size_t)rowA * DIM;
    __builtin_prefetch(xrow, 0, 0);                 // global_prefetch_b8

    v8f c[8] = {};                                  // 8 x (16x16 f32 accum)

    for (int k = 0; k < DIM; k += 4) {
        const int kk = k + 2 * half;                // half-wave holds K+2 slice
        const float2 xa = *(const float2*)(xrow + kk);
        const uint32_t bi = (uint32_t)rowA * DIM + (uint32_t)kk;
        v2f a;
        a.x = xa.x * drop_mask(bi,     (uint32_t)s);
        a.y = xa.y * drop_mask(bi + 1, (uint32_t)s);
#pragma unroll
        for (int j = 0; j < 8; ++j) {
            v2f bf;                                 // B: lanes = N cols
            bf.x = sW[kk * DIM + j * 16 + l16];
            bf.y = sW[(kk + 1) * DIM + j * 16 + l16];
            c[j] = __builtin_amdgcn_wmma_f32_16x16x4_f32(
                false, a, false, bf, (short)0, c[j], false, false);
        }
    }

    // C/D layout: VGPR r holds rows (tile*16 + r + half*8), col = j*16 + l16
    float dv[8];
#pragma unroll
    for (int r = 0; r < 8; ++r) dv[r] = dinv[tile * 16 + r + half * 8];
#pragma unroll
    for (int j = 0; j < 8; ++j) {
        const int col = j * 16 + l16;
#pragma unroll
        for (int r = 0; r < 8; ++r) {
            const int row = tile * 16 + r + half * 8;
            h[(size_t)row * DIM + col] = c[j][r] * dv[r];
        }
    }
}

// acc[i,:] = h'[i,:] * dinv[i] + b[:]   (self-loop term + bias)
__global__ void __launch_bounds__(256)
k_selfbias(const float* __restrict__ h, const float* __restrict__ dinv,
           const float* __restrict__ b, float* __restrict__ acc) {
    int i = blockIdx.x * blockDim.x + threadIdx.x;          // N*32 threads
    if (i >= NNODES * (DIM / 4)) return;
    const int node = i >> 5, q = i & 31;
    const float d = dinv[node];
    const float4 hv = ((const float4*)(h + (size_t)node * DIM))[q];
    const float4 bv = ((const float4*)b)[q];
    float4 o = make_float4(hv.x * d + bv.x, hv.y * d + bv.y,
                           hv.z * d + bv.z, hv.w * d + bv.w);
    ((float4*)(acc + (size_t)node * DIM))[q] = o;
}

// One wave32 per edge: acc[dst,:] += h'[src,:] * dinv[dst]
// h' and acc are both L2-resident (2 x 25.6 MB << 192 MB L2).
__global__ void __launch_bounds__(256)
k_scatter(const int* __restrict__ src, const int* __restrict__ dst,
          const float* __restrict__ h, const float* __restrict__ dinv,
          float* __restrict__ acc) {
    const int e = blockIdx.x * 8 + (threadIdx.x >> 5);
    if (e >= NEDGES) return;
    const int lane = threadIdx.x & 31;
    const int sn = src[e];
    const int dn = dst[e];
    const float w = dinv[dn];
    const float4 hv = *(const float4*)(h + (size_t)sn * DIM + lane * 4);
    float* ap = acc + (size_t)dn * DIM + lane * 4;
    atomicAdd(ap + 0, hv.x * w);
    atomicAdd(ap + 1, hv.y * w);
    atomicAdd(ap + 2, hv.z * w);
    atomicAdd(ap + 3, hv.w * w);
}

__global__ void __launch_bounds__(256)
k_accrelu(const float* __restrict__ acc, float* __restrict__ out) {
    int i = blockIdx.x * blockDim.x + threadIdx.x;          // N*DIM/4 threads
    if (i >= NNODES * DIM / 4) return;
    const float4 a = ((const float4*)acc)[i];
    float4 o = ((float4*)out)[i];
    o.x += fmaxf(a.x, 0.f); o.y += fmaxf(a.y, 0.f);
    o.z += fmaxf(a.z, 0.f); o.w += fmaxf(a.w, 0.f);
    ((float4*)out)[i] = o;
}

// Row-wise L2 normalize; one wave32 per 128-float row.
__global__ void __launch_bounds__(256)
k_normalize(float* __restrict__ out) {
    const int row = blockIdx.x * 8 + (threadIdx.x >> 5);
    if (row >= NNODES) return;
    const int lane = threadIdx.x & 31;
    float4 v = *(float4*)(out + (size_t)row * DIM + lane * 4);
    float ss = v.x * v.x + v.y * v.y + v.z * v.z + v.w * v.w;
#pragma unroll
    for (int m = 16; m > 0; m >>= 1) ss += __shfl_xor(ss, m, 32);
    const float sc = 1.0f / fmaxf(sqrtf(ss), 1e-12f);
    v.x *= sc; v.y *= sc; v.z *= sc; v.w *= sc;
    *(float4*)(out + (size_t)row * DIM + lane * 4) = v;
}

extern "C" void kernel_launch(void* const* d_in, const int* in_sizes, int n_in,
                              void* d_out, int out_size, void* d_ws, size_t ws_size,
                              hipStream_t stream) {
    const float* x  = (const float*)d_in[0];       // [N,128]
    const float* W  = (const float*)d_in[1];       // [S,128,128]
    const float* b  = (const float*)d_in[2];       // [S,128]
    const int*   ei = (const int*)d_in[3];         // [S,2,E]
    float* out = (float*)d_out;

    // workspace: dinv [N] | h' [N*128] | acc [N*128]  (~51.4 MB)
    float* degv = (float*)d_ws;
    float* hbuf = degv + NNODES;
    float* accb = hbuf + (size_t)NNODES * DIM;

    k_zero4<<<(NNODES * DIM / 4 + 255) / 256, 256, 0, stream>>>(out, NNODES * DIM / 4);

    for (int s = 0; s < NSUB; ++s) {
        const int* srcp = ei + (size_t)s * 2 * NEDGES;
        const int* dstp = srcp + NEDGES;
        const float* Ws = W + (size_t)s * DIM * DIM;
        const float* bs = b + (size_t)s * DIM;

        k_zero4<<<(NNODES / 4 + 255) / 256, 256, 0, stream>>>(degv, NNODES / 4);
        k_degree<<<(NEDGES + 255) / 256, 256, 0, stream>>>(dstp, degv, NEDGES);
        k_dinv<<<(NNODES + 255) / 256, 256, 0, stream>>>(degv, NNODES);
        k_gemm_wmma<<<(NTILES + 3) / 4, 128, 0, stream>>>(x, Ws, degv, hbuf, s);
        k_selfbias<<<(NNODES * (DIM / 4) + 255) / 256, 256, 0, stream>>>(hbuf, degv, bs, accb);
        k_scatter<<<(NEDGES + 7) / 8, 256, 0, stream>>>(srcp, dstp, hbuf, degv, accb);
        k_accrelu<<<(NNODES * DIM / 4 + 255) / 256, 256, 0, stream>>>(accb, out);
    }

    k_normalize<<<(NNODES + 7) / 8, 256, 0, stream>>>(out);
}